// DecoderRNN_11390253269479
// MI455X (gfx1250) — compile-verified
//
#include <hip/hip_runtime.h>
#include <cmath>

typedef __attribute__((ext_vector_type(2))) float v2f;
typedef __attribute__((ext_vector_type(8))) float v8f;

#define B_   128
#define T_   20
#define E_   512
#define H_   512
#define V_   10000
#define LDSS 516   // padded LDS row stride in floats (16 lanes -> distinct banks)

__device__ __forceinline__ float sigmoidf_(float x) {
    return 1.0f / (1.0f + __expf(-x));
}

__device__ __forceinline__ v8f wmma_f32(v2f a, v2f b, v8f c) {
    return __builtin_amdgcn_wmma_f32_16x16x4_f32(false, a, false, b, (short)0, c,
                                                 false, false);
}

// ---------------------------------------------------------------- init: h0 = features, c0 = 0
__global__ void lstm_init(const float* __restrict__ features,
                          float* __restrict__ h0, float* __restrict__ c0) {
    int i = blockIdx.x * blockDim.x + threadIdx.x;
    if (i < B_ * H_) { h0[i] = features[i]; c0[i] = 0.0f; }
}

// ---------------------------------------------------------------- fused embed + gates GEMM + LSTM cell
// grid (8,8): x = group of 4 H-tiles, y = M-tile. block = 128 threads (4 waves).
// Each wave owns 16 columns of H and accumulates all 4 gate tiles (N-stride 512),
// so the LSTM nonlinearity runs directly on the accumulator fragments.
__global__ __launch_bounds__(128) void lstm_step(
    const int* __restrict__ captions, int t,
    const float* __restrict__ embed_W,
    const float* __restrict__ W_ih, const float* __restrict__ W_hh,
    const float* __restrict__ b_ih, const float* __restrict__ b_hh,
    const float* __restrict__ h_in, const float* __restrict__ c_in,
    float* __restrict__ h_out, float* __restrict__ c_out)
{
    __shared__ float xt[16 * LDSS];             // 33 KB each; both staged up front,
    __shared__ float ht[16 * LDSS];             // single barrier, back-to-back K phases
    const int mtile = blockIdx.y;               // 0..7
    const int tid   = threadIdx.x;

    const int wave  = tid >> 5;                 // 0..3
    const int lane  = tid & 31;
    const int half  = lane >> 4;
    const int l16   = lane & 15;
    const int htile = blockIdx.x * 4 + wave;    // 0..31
    const int col   = htile * 16 + l16;         // 0..511 (column within each gate)

    v8f acc[4];
    #pragma unroll
    for (int g = 0; g < 4; ++g)
        #pragma unroll
        for (int v = 0; v < 8; ++v) acc[g][v] = 0.0f;

    // ---- stage x tile (embedding gather) and h tile: 16 rows x 512 each
    for (int idx = tid * 4; idx < 16 * E_; idx += 128 * 4) {
        int row = idx / E_, e = idx % E_;
        int tok = captions[(mtile * 16 + row) * T_ + t];
        *(float4*)(xt + row * LDSS + e) =
            *(const float4*)(embed_W + (size_t)tok * E_ + e);
        *(float4*)(ht + row * LDSS + e) =
            *(const float4*)(h_in + (size_t)(mtile * 16 + row) * H_ + e);
    }
    __syncthreads();

    // ---- phase 1: gates += x @ W_ih^T
    for (int k0 = 0; k0 < E_; k0 += 4) {
        int kk = k0 + 2 * half;
        v2f a = *(const v2f*)(xt + l16 * LDSS + kk);
        #pragma unroll
        for (int g = 0; g < 4; ++g) {
            v2f b = *(const v2f*)(W_ih + (size_t)(g * H_ + col) * E_ + kk);
            acc[g] = wmma_f32(a, b, acc[g]);
        }
    }
    // ---- phase 2: gates += h @ W_hh^T
    for (int k0 = 0; k0 < H_; k0 += 4) {
        int kk = k0 + 2 * half;
        v2f a = *(const v2f*)(ht + l16 * LDSS + kk);
        #pragma unroll
        for (int g = 0; g < 4; ++g) {
            v2f b = *(const v2f*)(W_hh + (size_t)(g * H_ + col) * H_ + kk);
            acc[g] = wmma_f32(a, b, acc[g]);
        }
    }

    // ---- LSTM cell on accumulator fragments (PyTorch gate order i,f,g,o)
    const float bi = b_ih[0 * H_ + col] + b_hh[0 * H_ + col];
    const float bf = b_ih[1 * H_ + col] + b_hh[1 * H_ + col];
    const float bg = b_ih[2 * H_ + col] + b_hh[2 * H_ + col];
    const float bo = b_ih[3 * H_ + col] + b_hh[3 * H_ + col];
    #pragma unroll
    for (int v = 0; v < 8; ++v) {
        int rowg = mtile * 16 + v + 8 * half;
        float iv = sigmoidf_(acc[0][v] + bi);
        float fv = sigmoidf_(acc[1][v] + bf);
        float gv = tanhf(acc[2][v] + bg);
        float ov = sigmoidf_(acc[3][v] + bo);
        float cp = c_in[(size_t)rowg * H_ + col];
        float cn = fv * cp + iv * gv;
        c_out[(size_t)rowg * H_ + col] = cn;
        h_out[(size_t)rowg * H_ + col] = ov * tanhf(cn);
    }
}

// ---------------------------------------------------------------- vocab projection: out[:,t,:] = h @ lin_W^T + lin_b
// grid (25,4): x = group of 25 N-tiles, y = pair of M-tiles. block = 160 threads (5 waves).
// Each wave: 5 N-tiles x 2 M-tiles -> 10 accumulators; every lin_W fragment feeds 2 WMMAs
// (halves L2 read traffic on the dominant 20.5 MB weight stream).
__global__ __launch_bounds__(160) void proj_step(
    const float* __restrict__ h, const float* __restrict__ lin_W,
    const float* __restrict__ lin_b, float* __restrict__ out, int t)
{
    __shared__ float hsh[32 * LDSS];            // 66 KB: 32 rows of h
    const int mbase = blockIdx.y * 32;          // first of 32 batch rows
    const int tid   = threadIdx.x;

    for (int idx = tid * 4; idx < 32 * H_; idx += 160 * 4) {
        int row = idx / H_, e = idx % H_;
        *(float4*)(hsh + row * LDSS + e) =
            *(const float4*)(h + (size_t)(mbase + row) * H_ + e);
    }
    __syncthreads();

    const int wave = tid >> 5;                    // 0..4
    const int lane = tid & 31;
    const int half = lane >> 4, l16 = lane & 15;
    const int ntile0 = blockIdx.x * 25 + wave * 5; // 625 tiles total (10000 = 625*16)

    v8f acc[10];                                   // [m][i] = acc[m*5+i]
    int  ncol[5];
    #pragma unroll
    for (int i = 0; i < 5; ++i) ncol[i] = (ntile0 + i) * 16 + l16;
    #pragma unroll
    for (int j = 0; j < 10; ++j)
        #pragma unroll
        for (int v = 0; v < 8; ++v) acc[j][v] = 0.0f;

    for (int k0 = 0; k0 < H_; k0 += 4) {
        int kk = k0 + 2 * half;
        v2f a0 = *(const v2f*)(hsh + l16 * LDSS + kk);          // M-tile 0
        v2f a1 = *(const v2f*)(hsh + (16 + l16) * LDSS + kk);   // M-tile 1
        #pragma unroll
        for (int i = 0; i < 5; ++i) {
            v2f b = *(const v2f*)(lin_W + (size_t)ncol[i] * H_ + kk);
            acc[i]     = wmma_f32(a0, b, acc[i]);
            acc[5 + i] = wmma_f32(a1, b, acc[5 + i]);
        }
    }

    #pragma unroll
    for (int m = 0; m < 2; ++m) {
        #pragma unroll
        for (int i = 0; i < 5; ++i) {
            float bias = lin_b[ncol[i]];
            #pragma unroll
            for (int v = 0; v < 8; ++v) {
                int rowg = mbase + m * 16 + v + 8 * half;
                // streaming store: write-once output, keep weights resident in L2
                __builtin_nontemporal_store(
                    acc[m * 5 + i][v] + bias,
                    out + (size_t)rowg * T_ * V_ + (size_t)t * V_ + ncol[i]);
            }
        }
    }
}

// ----------------------------------------------------------------
extern "C" void kernel_launch(void* const* d_in, const int* in_sizes, int n_in,
                              void* d_out, int out_size, void* d_ws, size_t ws_size,
                              hipStream_t stream) {
    const float* features = (const float*)d_in[0];
    const int*   captions = (const int*)d_in[1];
    const float* embed_W  = (const float*)d_in[2];
    const float* W_ih     = (const float*)d_in[3];
    const float* W_hh     = (const float*)d_in[4];
    const float* b_ih     = (const float*)d_in[5];
    const float* b_hh     = (const float*)d_in[6];
    const float* lin_W    = (const float*)d_in[7];
    const float* lin_b    = (const float*)d_in[8];
    float* out = (float*)d_out;

    // workspace: ping-pong h and c buffers (4 * 128*512 floats = 1 MB)
    float* ws = (float*)d_ws;
    float* hbuf[2] = { ws,                       ws + (size_t)B_ * H_ };
    float* cbuf[2] = { ws + 2 * (size_t)B_ * H_, ws + 3 * (size_t)B_ * H_ };

    lstm_init<<<(B_ * H_ + 255) / 256, 256, 0, stream>>>(features, hbuf[0], cbuf[0]);

    for (int t = 0; t < T_; ++t) {
        int cur = t & 1, nxt = cur ^ 1;
        lstm_step<<<dim3(8, 8), 128, 0, stream>>>(
            captions, t, embed_W, W_ih, W_hh, b_ih, b_hh,
            hbuf[cur], cbuf[cur], hbuf[nxt], cbuf[nxt]);
        proj_step<<<dim3(25, 4), 160, 0, stream>>>(hbuf[nxt], lin_W, lin_b, out, t);
    }
}